// RNAFeatures_74637941670408
// MI455X (gfx1250) — compile-verified
//
#include <hip/hip_runtime.h>
#include <hip/hip_bf16.h>
#include <float.h>

typedef __attribute__((ext_vector_type(2))) float v2f;
typedef __attribute__((ext_vector_type(8))) float v8f;
typedef __attribute__((ext_vector_type(4))) unsigned int v4u;
typedef __attribute__((ext_vector_type(8))) int v8i;
typedef __attribute__((ext_vector_type(4))) int v4i;

#define TOPK     30
#define NATOM    5
#define NRBF     32
#define PEDIM    16
#define KDIM     816        /* PEDIM + 25*NRBF */
#define EDGEDIM  128
#define FPAD     818        /* feats row stride (dwords): 818 % 64 = 50 -> no 4-way conflicts */
#define HSTR     132        /* padded LDS stride for h */

/* B staging via TDM: K-major ee_W, chunks of KC rows x 128 cols, LDS row
   stride 144 dwords (TDM pad: 16 dwords every 128) -> half-waves hit
   disjoint bank groups. */
#define KC        64
#define NCHUNK    13        /* ceil(816/64); last chunk = 48 rows */
#define BROW      144
#define BCHUNK_DW (KC * BROW)

/* LDS layout (dword offsets), dynamic shared only -> base offset 0 */
#define OFF_FEATS 0
#define OFF_HBUF  (32 * FPAD)                 /* 26176 */
#define OFF_BBUF  (OFF_HBUF + 32 * HSTR)      /* 30400 */
#define OFF_EIDX  (OFF_BBUF + 2 * BCHUNK_DW)  /* 48832 */
#define OFF_XI    (OFF_EIDX + 32)             /* 48864 */
#define SMEM_DW   (OFF_XI + 16)               /* 48880 dwords = 195520 B */

// ---------------------------------------------------------------------------
// Kernel 1: per-row masked distances + iterative top-30 selection.
// ---------------------------------------------------------------------------
__global__ __launch_bounds__(256) void rnafeat_topk_kernel(
    const float* __restrict__ xyz, const float* __restrict__ mask,
    float* __restrict__ d_nb, int* __restrict__ e_idx, int n)
{
    __shared__ float dv[512];
    __shared__ float da[512];
    __shared__ float rv[256];
    __shared__ int   ri[256];
    __shared__ float xi[3];
    __shared__ float s_dmax;

    const int i   = blockIdx.x;
    const int tid = threadIdx.x;

    if (tid < 3) xi[tid] = xyz[i * NATOM * 3 + tid];
    __syncthreads();

    const float mi = mask[i];
    float lmax = -FLT_MAX;
    for (int j = tid; j < n; j += 256) {
        float dx = xi[0] - xyz[j * NATOM * 3 + 0];
        float dy = xi[1] - xyz[j * NATOM * 3 + 1];
        float dz = xi[2] - xyz[j * NATOM * 3 + 2];
        float m2 = mi * mask[j];
        float d  = sqrtf(dx * dx + dy * dy + dz * dz + 1e-6f) * m2;
        dv[j] = d;
        lmax = fmaxf(lmax, d);
    }
    rv[tid] = lmax;
    __syncthreads();
    for (int s = 128; s > 0; s >>= 1) {
        if (tid < s) rv[tid] = fmaxf(rv[tid], rv[tid + s]);
        __syncthreads();
    }
    if (tid == 0) s_dmax = rv[0];
    __syncthreads();
    const float dmax = s_dmax;
    for (int j = tid; j < n; j += 256) {
        float m2 = mi * mask[j];
        da[j] = dv[j] + (1.0f - m2) * dmax;
    }
    __syncthreads();

    for (int k = 0; k < TOPK; ++k) {
        float bv = FLT_MAX;
        int   bi = 0x7fffffff;
        for (int j = tid; j < n; j += 256) {
            float v = da[j];
            if (v < bv || (v == bv && j < bi)) { bv = v; bi = j; }
        }
        rv[tid] = bv; ri[tid] = bi;
        __syncthreads();
        for (int s = 128; s > 0; s >>= 1) {
            if (tid < s) {
                float ov = rv[tid + s]; int oi = ri[tid + s];
                if (ov < rv[tid] || (ov == rv[tid] && oi < ri[tid])) {
                    rv[tid] = ov; ri[tid] = oi;
                }
            }
            __syncthreads();
        }
        if (tid == 0) {
            int idx = ri[0];
            e_idx[i * TOPK + k] = idx;
            d_nb[i * TOPK + k]  = rv[0];
            da[idx] = FLT_MAX;
        }
        __syncthreads();
    }
}

// ---------------------------------------------------------------------------
// Transpose ee_W (128 x 816, row-major) -> eeWT (816 x 128, K-major) in d_ws.
// ---------------------------------------------------------------------------
__global__ __launch_bounds__(256) void rnafeat_transpose_kernel(
    const float* __restrict__ w, float* __restrict__ wt)
{
    int idx = blockIdx.x * 256 + threadIdx.x;       // 0 .. 816*128-1
    if (idx < KDIM * EDGEDIM) {
        int k = idx >> 7;
        int nn = idx & (EDGEDIM - 1);
        wt[idx] = w[nn * KDIM + k];
    }
}

// ---------------------------------------------------------------------------
// TDM: DMA `nelems` dwords from gsrc into LDS at byte offset lds_off,
// inserting 16 pad dwords every 128 dwords (-> 144-dword LDS row stride).
// Descriptor per CDNA5 ISA ch.8: 1-D tile, tensor_dim0 == tile_dim0.
// Uses the 6-arg (clang-23 / therock) builtin form.
// ---------------------------------------------------------------------------
__device__ __forceinline__ void tdm_load_dwords(const float* gsrc,
                                                unsigned lds_off_bytes,
                                                unsigned nelems)
{
    unsigned long long ga = (unsigned long long)(const void*)gsrc;
    v4u g0;
    g0[0] = 1u;                                   // count=1, user mode, no gather
    g0[1] = lds_off_bytes;                        // lds_addr
    g0[2] = (unsigned)(ga & 0xFFFFFFFFu);         // global_addr[31:0]
    g0[3] = (unsigned)((ga >> 32) & 0x01FFFFFFu)  // global_addr[56:32]
          | (2u << 30);                           // type=2 ("image")
    v8i g1;
    g1[0] = (int)((2u << 16)        // data_size = 4B
          |      (1u << 20)         // pad_enable
          |      (6u << 22)         // pad_interval: 128 dwords
          |      (15u << 25));      // pad_amount: 16 dwords
    g1[1] = (int)((nelems & 0xFFFFu) << 16);          // tensor_dim0[15:0]
    g1[2] = (int)(((nelems >> 16) & 0xFFFFu)          // tensor_dim0[31:16]
          |      (1u << 16));                         // tensor_dim1 = 1
    g1[3] = (int)((nelems & 0xFFFFu) << 16);          // tile_dim0 = nelems
    g1[4] = 0;                                        // tile_dim1/2 unused
    g1[5] = (int)nelems;                              // tensor_dim0_stride lo32
    g1[6] = 0;
    g1[7] = 0;
    v4i z4 = {0, 0, 0, 0};
    v8i z8 = {0, 0, 0, 0, 0, 0, 0, 0};
    __builtin_amdgcn_tensor_load_to_lds(g0, g1, z4, z4, z8, 0);
}

// ---------------------------------------------------------------------------
// Kernel 2: features + fp32 WMMA GEMM (B streamed by TDM) + LayerNorm.
// 128 threads = 4 waves; wave owns 2 of 8 N-tiles and DMAs 1/4 of each
// B chunk. Chunk 0 DMA overlaps the transcendental feature phase.
// ---------------------------------------------------------------------------
__global__ __launch_bounds__(128) void rnafeat_edge_kernel(
    const float* __restrict__ xyz,
    const int*   __restrict__ chain_idx,
    const int*   __restrict__ residue_idx,
    const float* __restrict__ pe_W,  const float* __restrict__ pe_b,
    const float* __restrict__ eeWT,                 /* (816,128) K-major */
    const float* __restrict__ ln_g,  const float* __restrict__ ln_b,
    const int*   __restrict__ e_idx,
    float* __restrict__ out)
{
    extern __shared__ float smem[];
    float* feats  = smem + OFF_FEATS;
    float* hbuf   = smem + OFF_HBUF;
    int*   eidx_s = (int*)(smem + OFF_EIDX);
    float* xi_s   = smem + OFF_XI;

    const int i    = blockIdx.x;
    const int tid  = threadIdx.x;
    const int wave = tid >> 5;
    const int lane = tid & 31;

    // ---- kick off B chunk 0 DMA (each wave copies its quarter) ----
    {
        int rpw = KC >> 2;                         // 16 rows per wave
        int r0  = rpw * wave;
        tdm_load_dwords(eeWT + (size_t)r0 * EDGEDIM,
                        (unsigned)(OFF_BBUF + r0 * BROW) * 4u,
                        (unsigned)(rpw * EDGEDIM));
    }

    if (tid < TOPK) eidx_s[tid] = e_idx[i * TOPK + tid];
    if (tid < NATOM * 3) xi_s[tid] = xyz[i * NATOM * 3 + tid];
    for (int idx = tid; idx < 2 * FPAD; idx += 128) feats[30 * FPAD + idx] = 0.0f;
    __syncthreads();

    const int ic = chain_idx[i];
    const int ir = residue_idx[i];

    // ---- feature build: one neighbor per wave per iteration ----
    const float sigma  = 20.0f / 32.0f;
    const float inv2s2 = 1.0f / (2.0f * sigma * sigma);
    const float miu    = (float)(lane + 1) * sigma;

    for (int it = 0; it < 8; ++it) {
        int jn = wave + 4 * it;
        if (jn >= TOPK) break;
        int jj = eidx_s[jn];

        float distp = 0.0f;
        if (lane < 25) {
            int a = lane / 5, b = lane - 5 * a;
            float dx = xi_s[a * 3 + 0] - xyz[jj * NATOM * 3 + b * 3 + 0];
            float dy = xi_s[a * 3 + 1] - xyz[jj * NATOM * 3 + b * 3 + 1];
            float dz = xi_s[a * 3 + 2] - xyz[jj * NATOM * 3 + b * 3 + 2];
            distp = sqrtf(dx * dx + dy * dy + dz * dz);
        }
        if (lane < PEDIM) {
            int off = ir - residue_idx[jj];
            int sc  = (ic == chain_idx[jj]);
            int dcl = off + 32;
            dcl = dcl < 0 ? 0 : (dcl > 64 ? 64 : dcl);
            dcl = sc ? dcl : 65;
            feats[jn * FPAD + lane] = pe_W[lane * 66 + dcl] + pe_b[lane];
        }
        #pragma unroll
        for (int p = 0; p < 25; ++p) {
            float dp = __shfl(distp, p, 32);
            float t  = dp - miu;
            feats[jn * FPAD + PEDIM + p * NRBF + lane] = __expf(-t * t * inv2s2);
        }
    }
    __syncthreads();   // feats ready (chunk-0 DMA still in flight, overlapped)

    // ---- WMMA GEMM: h(32x128) = feats(32x816) x ee_W^T, K-chunked ----
    const int hi  = lane >> 4;
    const int l15 = lane & 15;
    const int nb  = wave * 32;

    v8f acc00 = {}; v8f acc01 = {}; v8f acc10 = {}; v8f acc11 = {};
    const float* arow0 = &feats[l15 * FPAD];
    const float* arow1 = &feats[(16 + l15) * FPAD];

    for (int c = 0; c < NCHUNK; ++c) {
        const int cbase = c * KC;
        const int kc    = (KDIM - cbase) < KC ? (KDIM - cbase) : KC;

        if (c + 1 < NCHUNK) {                     // prefetch next chunk
            int nbase = (c + 1) * KC;
            int nkc   = (KDIM - nbase) < KC ? (KDIM - nbase) : KC;
            int rpw   = nkc >> 2;
            int r0    = rpw * wave;
            tdm_load_dwords(eeWT + (size_t)(nbase + r0) * EDGEDIM,
                            (unsigned)(OFF_BBUF + ((c + 1) & 1) * BCHUNK_DW
                                       + r0 * BROW) * 4u,
                            (unsigned)(rpw * EDGEDIM));
            __builtin_amdgcn_s_wait_tensorcnt(1); // chunk c complete (in-order)
        } else {
            __builtin_amdgcn_s_wait_tensorcnt(0);
        }
        __syncthreads();                          // all quarters of chunk c in LDS

        const float* bb = smem + OFF_BBUF + (c & 1) * BCHUNK_DW;
        #pragma unroll 2
        for (int kl = 0; kl < kc; kl += 4) {
            int ko = cbase + kl + 2 * hi;         // absolute K for A
            int r  = kl + 2 * hi;                 // chunk-local K row for B
            v2f a0 = *(const v2f*)(arow0 + ko);
            v2f a1 = *(const v2f*)(arow1 + ko);
            v2f b0 = { bb[r * BROW + nb + l15],      bb[(r + 1) * BROW + nb + l15] };
            v2f b1 = { bb[r * BROW + nb + 16 + l15], bb[(r + 1) * BROW + nb + 16 + l15] };
            acc00 = __builtin_amdgcn_wmma_f32_16x16x4_f32(false, a0, false, b0,
                                                          (short)0, acc00, false, false);
            acc01 = __builtin_amdgcn_wmma_f32_16x16x4_f32(false, a0, false, b1,
                                                          (short)0, acc01, false, false);
            acc10 = __builtin_amdgcn_wmma_f32_16x16x4_f32(false, a1, false, b0,
                                                          (short)0, acc10, false, false);
            acc11 = __builtin_amdgcn_wmma_f32_16x16x4_f32(false, a1, false, b1,
                                                          (short)0, acc11, false, false);
        }
        __syncthreads();      // chunk c fully consumed -> its buffer reusable
    }

    // C/D layout: VGPR v -> row v + 8*hi, col = l15 (+ tile offsets)
    {
        const int r0 = 8 * hi;
        const int c0 = nb + l15;
        #pragma unroll
        for (int v = 0; v < 8; ++v) {
            hbuf[(r0 + v) * HSTR + c0]           = acc00[v];
            hbuf[(r0 + v) * HSTR + c0 + 16]      = acc01[v];
            hbuf[(16 + r0 + v) * HSTR + c0]      = acc10[v];
            hbuf[(16 + r0 + v) * HSTR + c0 + 16] = acc11[v];
        }
    }
    __syncthreads();

    // ---- LayerNorm over 128 per row; wave handles 8 rows via shuffles ----
    for (int q = 0; q < 8; ++q) {
        int r = wave * 8 + q;
        if (r >= TOPK) break;
        float v0 = hbuf[r * HSTR + lane];
        float v1 = hbuf[r * HSTR + lane + 32];
        float v2 = hbuf[r * HSTR + lane + 64];
        float v3 = hbuf[r * HSTR + lane + 96];
        float s  = v0 + v1 + v2 + v3;
        float s2 = v0 * v0 + v1 * v1 + v2 * v2 + v3 * v3;
        #pragma unroll
        for (int m = 16; m >= 1; m >>= 1) {
            s  += __shfl_xor(s,  m, 32);
            s2 += __shfl_xor(s2, m, 32);
        }
        float mu  = s * (1.0f / 128.0f);
        float var = s2 * (1.0f / 128.0f) - mu * mu;
        float rs  = rsqrtf(var + 1e-5f);
        float* orow = out + ((size_t)i * TOPK + r) * EDGEDIM;
        orow[lane]      = (v0 - mu) * rs * ln_g[lane]      + ln_b[lane];
        orow[lane + 32] = (v1 - mu) * rs * ln_g[lane + 32] + ln_b[lane + 32];
        orow[lane + 64] = (v2 - mu) * rs * ln_g[lane + 64] + ln_b[lane + 64];
        orow[lane + 96] = (v3 - mu) * rs * ln_g[lane + 96] + ln_b[lane + 96];
    }
}

extern "C" void kernel_launch(void* const* d_in, const int* in_sizes, int n_in,
                              void* d_out, int out_size, void* d_ws, size_t ws_size,
                              hipStream_t stream) {
    const float* xyz         = (const float*)d_in[0];
    const float* mask        = (const float*)d_in[1];
    const int*   chain_idx   = (const int*)d_in[2];
    const int*   residue_idx = (const int*)d_in[3];
    const float* pe_W        = (const float*)d_in[4];
    const float* pe_b        = (const float*)d_in[5];
    const float* ee_W        = (const float*)d_in[6];
    const float* ln_g        = (const float*)d_in[7];
    const float* ln_b        = (const float*)d_in[8];

    const int n = in_sizes[1];   // L

    float* out  = (float*)d_out;                              // (L, 30, 128)
    float* d_nb = out + (size_t)n * TOPK * EDGEDIM;           // (L, 30)
    int*   eidx = (int*)(d_nb + (size_t)n * TOPK);            // (L, 30) int32
    float* eeWT = (float*)d_ws;                               // (816, 128)

    rnafeat_topk_kernel<<<n, 256, 0, stream>>>(xyz, mask, d_nb, eidx, n);

    rnafeat_transpose_kernel<<<(KDIM * EDGEDIM + 255) / 256, 256, 0, stream>>>(
        ee_W, eeWT);

    rnafeat_edge_kernel<<<n, 128, SMEM_DW * sizeof(float), stream>>>(
        xyz, chain_idx, residue_idx, pe_W, pe_b, eeWT, ln_g, ln_b, eidx, out);
}